// PromptedGAT_54417235640800
// MI455X (gfx1250) — compile-verified
//
#include <hip/hip_runtime.h>
#include <math.h>

typedef __attribute__((ext_vector_type(2))) float v2f;
typedef __attribute__((ext_vector_type(8))) float v8f;

#define GAT_N_NODES 100000
#define GAT_N_EDGES 1600000
#define GAT_IN_FEATS 128
#define GAT_H_FEATS 64
#define GAT_NEG_SLOPE 0.2f

// ---------------------------------------------------------------------------
// GEMM: H[n x 64] = act(X[n x K]) @ W[K x 64], fp32 WMMA 16x16x4.
// Block = 128 threads = 4 waves; each wave computes a 16-row x 64-col tile.
// A layout (16x4 f32): lanes 0-15 -> M=lane, VGPR0=K0,VGPR1=K1;
//                      lanes 16-31 -> M=lane-16, VGPR0=K2,VGPR1=K3.
// B layout (4x16 f32, mirrored): lanes 0-15 -> N=lane, K pair {0,1};
//                                lanes 16-31 -> N=lane-16, K pair {2,3}.
// C/D layout (16x16 f32): lanes 0-15 -> M=j,N=lane; lanes 16-31 -> M=8+j.
// Out-of-range rows are CLAMPED (not predicated) on load so the inner loop
// carries no EXEC manipulation; the D-store is row-guarded instead.
// ---------------------------------------------------------------------------
template <bool RELU_IN>
__global__ void gat_gemm_wmma(const float* __restrict__ X,
                              const float* __restrict__ W,
                              float* __restrict__ H,
                              int n, int K) {
    const int lane = threadIdx.x & 31;
    const int wave = threadIdx.x >> 5;
    const int row0 = blockIdx.x * 64 + wave * 16;
    const int m  = lane & 15;   // row (A) / col (B,C) within tile
    const int kh = lane >> 4;   // selects K pair {0,1} vs {2,3}; M-half for C/D

    v8f acc[4];
    #pragma unroll
    for (int t = 0; t < 4; ++t) acc[t] = (v8f){0.f,0.f,0.f,0.f,0.f,0.f,0.f,0.f};

    const int r0 = row0 + m;
    const int r  = (r0 < n) ? r0 : (n - 1);   // clamp: garbage rows never stored
    const size_t rbase = (size_t)r * (size_t)K;

    for (int k0 = 0; k0 < K; k0 += 4) {
        const int kk = k0 + kh * 2;
        float a0 = X[rbase + kk];
        float a1 = X[rbase + kk + 1];
        if (RELU_IN) { a0 = fmaxf(a0, 0.f); a1 = fmaxf(a1, 0.f); }
        v2f A; A.x = a0; A.y = a1;

        #pragma unroll
        for (int t = 0; t < 4; ++t) {
            const int col = t * 16 + m;
            v2f B;
            B.x = W[(size_t)kk * GAT_H_FEATS + col];
            B.y = W[(size_t)(kk + 1) * GAT_H_FEATS + col];
            acc[t] = __builtin_amdgcn_wmma_f32_16x16x4_f32(
                false, A, false, B, (short)0, acc[t], false, false);
        }
    }

    #pragma unroll
    for (int j = 0; j < 8; ++j) {
        const int rr = row0 + kh * 8 + j;
        if (rr < n) {
            float* __restrict__ hrow = H + (size_t)rr * GAT_H_FEATS;
            hrow[ 0 + m] = acc[0][j];
            hrow[16 + m] = acc[1][j];
            hrow[32 + m] = acc[2][j];
            hrow[48 + m] = acc[3][j];
        }
    }
}

// el[i] = dot(H[i], attn_l), er[i] = dot(H[i], attn_r)
__global__ void gat_attn_scores(const float* __restrict__ H,
                                const float* __restrict__ al,
                                const float* __restrict__ ar,
                                float* __restrict__ el,
                                float* __restrict__ er, int n) {
    const int i = blockIdx.x * blockDim.x + threadIdx.x;
    if (i >= n) return;
    const float* __restrict__ h = H + (size_t)i * GAT_H_FEATS;
    float l = 0.f, r = 0.f;
    #pragma unroll
    for (int f = 0; f < GAT_H_FEATS; f += 4) {
        const float4 v = *(const float4*)(h + f);
        l += v.x * al[f]     + v.y * al[f + 1] + v.z * al[f + 2] + v.w * al[f + 3];
        r += v.x * ar[f]     + v.y * ar[f + 1] + v.z * ar[f + 2] + v.w * ar[f + 3];
    }
    el[i] = l;
    er[i] = r;
}

__global__ void gat_fill_f32(float* __restrict__ p, float v, int count) {
    const int i = blockIdx.x * blockDim.x + threadIdx.x;
    if (i < count) p[i] = v;
}

__device__ __forceinline__ void atomicMaxF32(float* addr, float v) {
    // IEEE ordering trick: ints compare like floats for non-negative values;
    // negative floats reverse-order as unsigned ints.
    if (v >= 0.f) atomicMax((int*)addr, __float_as_int(v));
    else          atomicMin((unsigned int*)addr, (unsigned int)__float_as_int(v));
}

// e = leaky_relu(el[src]+er[dst]); stash e; m[dst] = max(m[dst], e)
__global__ void gat_edge_score_max(const int* __restrict__ src,
                                   const int* __restrict__ dst,
                                   const float* __restrict__ el,
                                   const float* __restrict__ er,
                                   float* __restrict__ esc,
                                   float* __restrict__ mbuf, int E) {
    const int e = blockIdx.x * blockDim.x + threadIdx.x;
    if (e >= E) return;
    const int s = src[e];
    const int d = dst[e];
    float v = el[s] + er[d];
    v = (v >= 0.f) ? v : GAT_NEG_SLOPE * v;
    esc[e] = v;
    atomicMaxF32(&mbuf[d], v);
}

// esc[e] <- exp(esc[e]-m[dst]); z[dst] += esc[e]   (in-place score -> weight)
__global__ void gat_edge_exp_sum(const int* __restrict__ dst,
                                 float* __restrict__ esc,
                                 const float* __restrict__ mbuf,
                                 float* __restrict__ zbuf, int E) {
    const int e = blockIdx.x * blockDim.x + threadIdx.x;
    if (e >= E) return;
    const int d = dst[e];
    const float w = __expf(esc[e] - mbuf[d]);
    esc[e] = w;
    atomicAdd(&zbuf[d], w);
}

// One wave per edge; each lane handles 2 features.
// out[dst] += (ee/z[dst]) * H[src]
__global__ void gat_edge_aggregate(const int* __restrict__ src,
                                   const int* __restrict__ dst,
                                   const float* __restrict__ ee,
                                   const float* __restrict__ zbuf,
                                   const float* __restrict__ H,
                                   float* __restrict__ out, int E) {
    const int gid  = blockIdx.x * blockDim.x + threadIdx.x;
    const int e    = gid >> 5;
    const int lane = gid & 31;
    if (e >= E) return;
    const int s = src[e];
    const int d = dst[e];
    float z = zbuf[d];
    z = (z == 0.f) ? 1.f : z;
    const float alpha = ee[e] / z;
    const float2 hv = *(const float2*)(H + (size_t)s * GAT_H_FEATS + 2 * lane);
    float* __restrict__ o = out + (size_t)d * GAT_H_FEATS + 2 * lane;
    atomicAdd(o + 0, alpha * hv.x);
    atomicAdd(o + 1, alpha * hv.y);
}

__global__ void gat_relu_inplace(float* __restrict__ p, int count) {
    const int i = blockIdx.x * blockDim.x + threadIdx.x;
    if (i < count) p[i] = fmaxf(p[i], 0.f);
}

extern "C" void kernel_launch(void* const* d_in, const int* in_sizes, int n_in,
                              void* d_out, int out_size, void* d_ws, size_t ws_size,
                              hipStream_t stream) {
    (void)in_sizes; (void)n_in; (void)out_size; (void)ws_size;

    const float* feat = (const float*)d_in[0];
    const float* W1   = (const float*)d_in[1];
    const float* al1  = (const float*)d_in[2];
    const float* ar1  = (const float*)d_in[3];
    const float* W2   = (const float*)d_in[4];
    const float* al2  = (const float*)d_in[5];
    const float* ar2  = (const float*)d_in[6];
    const int*   src  = (const int*)d_in[7];
    const int*   dst  = (const int*)d_in[8];
    float*       out  = (float*)d_out;

    const int N = GAT_N_NODES;
    const int E = GAT_N_EDGES;
    const int F = GAT_H_FEATS;

    // Workspace layout (floats), ~59 MB total.
    char* ws = (char*)d_ws;
    float* H    = (float*)ws;  ws += (size_t)N * F * sizeof(float);  // post-GEMM feats
    float* agg  = (float*)ws;  ws += (size_t)N * F * sizeof(float);  // layer-1 output
    float* el   = (float*)ws;  ws += (size_t)N * sizeof(float);
    float* er   = (float*)ws;  ws += (size_t)N * sizeof(float);
    float* mbuf = (float*)ws;  ws += (size_t)N * sizeof(float);
    float* zbuf = (float*)ws;  ws += (size_t)N * sizeof(float);
    float* esc  = (float*)ws;  ws += (size_t)E * sizeof(float);      // scores -> weights

    const int TB = 256;
    const dim3 blkGemm(128);
    const dim3 grdGemm((N + 63) / 64);
    const int grdN    = (N + TB - 1) / TB;
    const int grdNF   = (N * F + TB - 1) / TB;
    const int grdE    = (E + TB - 1) / TB;
    const int grdE32  = (int)(((long long)E * 32 + TB - 1) / TB);

    // ---------------- Layer 1 ----------------
    gat_gemm_wmma<false><<<grdGemm, blkGemm, 0, stream>>>(feat, W1, H, N, GAT_IN_FEATS);
    gat_attn_scores<<<grdN, TB, 0, stream>>>(H, al1, ar1, el, er, N);
    gat_fill_f32<<<grdN,  TB, 0, stream>>>(mbuf, -INFINITY, N);
    gat_fill_f32<<<grdN,  TB, 0, stream>>>(zbuf, 0.f, N);
    gat_fill_f32<<<grdNF, TB, 0, stream>>>(agg, 0.f, N * F);
    gat_edge_score_max<<<grdE, TB, 0, stream>>>(src, dst, el, er, esc, mbuf, E);
    gat_edge_exp_sum<<<grdE, TB, 0, stream>>>(dst, esc, mbuf, zbuf, E);
    gat_edge_aggregate<<<grdE32, TB, 0, stream>>>(src, dst, esc, zbuf, H, agg, E);

    // ---------------- Layer 2 (ReLU folded into GEMM A-load) ----------------
    gat_gemm_wmma<true><<<grdGemm, blkGemm, 0, stream>>>(agg, W2, H, N, GAT_H_FEATS);
    gat_attn_scores<<<grdN, TB, 0, stream>>>(H, al2, ar2, el, er, N);
    gat_fill_f32<<<grdN,  TB, 0, stream>>>(mbuf, -INFINITY, N);
    gat_fill_f32<<<grdN,  TB, 0, stream>>>(zbuf, 0.f, N);
    gat_fill_f32<<<grdNF, TB, 0, stream>>>(out, 0.f, N * F);
    gat_edge_score_max<<<grdE, TB, 0, stream>>>(src, dst, el, er, esc, mbuf, E);
    gat_edge_exp_sum<<<grdE, TB, 0, stream>>>(dst, esc, mbuf, zbuf, E);
    gat_edge_aggregate<<<grdE32, TB, 0, stream>>>(src, dst, esc, zbuf, H, out, E);
    gat_relu_inplace<<<grdNF, TB, 0, stream>>>(out, N * F);
}